// AxisAlignConv_27522150432911
// MI455X (gfx1250) — compile-verified
//
#include <hip/hip_runtime.h>

#define KSX 3
#define KK 9
#define C_IN 256
#define O_OUT 256
#define Hd 64
#define Wd 64
#define HWd 4096
#define GROUPS 8
#define EPSV 1e-5f
#define MTILE 32
#define KCHUNK 32
#define NCHUNKS (KK * C_IN / KCHUNK)   // 72
#define CPAD 36                        // padded LDS row for epilogue transpose

typedef float v2f __attribute__((ext_vector_type(2)));
typedef float v8f __attribute__((ext_vector_type(8)));
typedef int   vi4 __attribute__((vector_size(16)));    // int4 payload type for b128

#define AS1 __attribute__((address_space(1)))
#define AS3 __attribute__((address_space(3)))

#if __has_builtin(__builtin_amdgcn_global_load_async_to_lds_b128)
#define HAVE_ASYNC_LDS 1
#else
#define HAVE_ASYNC_LDS 0
#endif

static __device__ __forceinline__ void async_wait0() {
#if HAVE_ASYNC_LDS
#if __has_builtin(__builtin_amdgcn_s_wait_asynccnt)
    __builtin_amdgcn_s_wait_asynccnt(0);
#else
    asm volatile("s_wait_asynccnt 0x0" ::: "memory");
#endif
#endif
}

// ---------------- kernel 0: weight transpose into pair-interleaved layout ----
// weight[o][c][k] -> wTp[(kc>>1)*512 + o*2 + (kc&1)],  kc = k*256 + c
// so {B[kc][o], B[kc+1][o]} sit adjacent -> one ds_load_b64 per B fragment.
__global__ __launch_bounds__(256) void wt_transpose_k(const float* __restrict__ w,
                                                      float* __restrict__ wTp) {
    int kc = blockIdx.x;           // 0..2303 == k*256 + c
    int k = kc >> 8;
    int c = kc & 255;
    int o = threadIdx.x;
    wTp[(kc >> 1) * (2 * O_OUT) + o * 2 + (kc & 1)] = w[(o * C_IN + c) * KK + k];
}

// ---------------- kernel 1: fused sample + GEMM (WMMA f32, double-buffered) --
__global__ __launch_bounds__(256) void conv_wmma_k(const float* __restrict__ x,
                                                   const float* __restrict__ angle,
                                                   const float* __restrict__ mask,
                                                   const float* __restrict__ wTp,
                                                   float* __restrict__ out) {
    __shared__ float As[2][MTILE][KCHUNK + 2];                  // rows 8B-aligned
    __shared__ __align__(16) float Bs[2][KCHUNK / 2][O_OUT][2]; // pair-interleaved
    __shared__ __align__(16) int   soff[MTILE * KK][4];         // bilinear corner offsets
    __shared__ __align__(16) float sqw[MTILE * KK][4];          // corner weights (mask folded)

    const int tid  = threadIdx.x;
    const int pix0 = blockIdx.x * MTILE;   // 32 consecutive pixels: same b, same h
    const int b    = pix0 >> 12;
    const int hw0  = pix0 & 4095;
    const int h    = hw0 >> 6;
    const int w0   = hw0 & 63;

    // ---- setup: per (pixel, tap) bilinear corners & weights (mask+validity folded)
    for (int t = tid; t < MTILE * KK; t += 256) {
        int m = t / KK;
        int k = t - m * KK;
        int wc = w0 + m;
        float ang = angle[b * HWd + h * Wd + wc];
        float sn, cs;
        __sincosf(ang, &sn, &cs);
        int ki = k / 3, kj = k - 3 * ki;
        float c0 = (float)(ki - 1);
        float c1 = (float)(kj - 1);
        float d0 = (cs - 1.0f) * c0 + sn * c1;
        float d1 = -sn * c0 + (cs - 1.0f) * c1;
        float py = (float)(h - 1 + ki) + d0;
        float px = (float)(wc - 1 + kj) + d1;
        float fy = floorf(py), fx = floorf(px);
        float wy = py - fy, wx = px - fx;
        int y0 = (int)fy, x0 = (int)fx;
        int y1 = y0 + 1, x1 = x0 + 1;
        float mv = mask[(b * KK + k) * HWd + h * Wd + wc];
        float vy0 = (y0 >= 0 && y0 < Hd) ? 1.0f : 0.0f;
        float vy1 = (y1 >= 0 && y1 < Hd) ? 1.0f : 0.0f;
        float vx0 = (x0 >= 0 && x0 < Wd) ? 1.0f : 0.0f;
        float vx1 = (x1 >= 0 && x1 < Wd) ? 1.0f : 0.0f;
        int y0c = min(max(y0, 0), Hd - 1), y1c = min(max(y1, 0), Hd - 1);
        int x0c = min(max(x0, 0), Wd - 1), x1c = min(max(x1, 0), Wd - 1);
        soff[t][0] = y0c * Wd + x0c;
        soff[t][1] = y0c * Wd + x1c;
        soff[t][2] = y1c * Wd + x0c;
        soff[t][3] = y1c * Wd + x1c;
        sqw[t][0] = (1.0f - wy) * (1.0f - wx) * mv * vy0 * vx0;
        sqw[t][1] = (1.0f - wy) * wx          * mv * vy0 * vx1;
        sqw[t][2] = wy          * (1.0f - wx) * mv * vy1 * vx0;
        sqw[t][3] = wy          * wx          * mv * vy1 * vx1;
    }

    const int wv    = tid >> 5;            // wave 0..7
    const int lane  = tid & 31;
    const int Ms    = (wv >> 2) * 16;      // M sub-tile base (0 or 16)
    const int Nw    = (wv & 3) * 64;       // N group base (0/64/128/192)
    const int lrow  = lane & 15;
    const int khalf = (lane >> 4) << 1;    // 0 or 2 (A/B K-split across half-waves)
    const int mpix  = tid & 31;            // this thread's fixed pixel for A-build

    // ---- per-thread cached bilinear params (refreshed once per tap)
    int   po[4];
    float pq[4];
    auto loadParams = [&](int k) {
        int t = mpix * KK + k;
#pragma unroll
        for (int q = 0; q < 4; ++q) {
            po[q] = soff[t][q];
            pq[q] = sqw[t][q];
        }
    };

    // ---- stage B chunk k rows 0..7 (8x256 fp32 == 512 x b128) into Bs[buf]
    auto stageB = [&](int ch, int buf) {
        const float* src = wTp + (size_t)ch * (KCHUNK * O_OUT);  // chunk = 8192 floats
#if HAVE_ASYNC_LDS
#pragma unroll
        for (int i = 0; i < 2; ++i) {
            int e = i * 256 + tid;         // b128 index 0..511
            float* sg       = (float*)(src + e * 4);             // drop const
            AS1 float* sg1  = (AS1 float*)sg;                    // -> global AS
            float* dl       = &Bs[buf][0][0][0] + e * 4;
            AS3 float* dl3  = (AS3 float*)dl;                    // -> LDS AS
            __builtin_amdgcn_global_load_async_to_lds_b128(
                (AS1 vi4*)sg1, (AS3 vi4*)dl3, 0, 0);
        }
#else
        const float4* s4 = (const float4*)src;
        float4* d4 = (float4*)&Bs[buf][0][0][0];
#pragma unroll
        for (int i = 0; i < 2; ++i)
            d4[i * 256 + tid] = s4[i * 256 + tid];
#endif
    };

    // ---- build A chunk (32 px x 32 ch): masked bilinear gather into As[buf]
    // thread's pixel fixed (mpix); wave covers one channel column per step -> coalesced
    auto buildA = [&](int ch, int buf) {
        const int cc = (ch & 7) * KCHUNK;  // channel base
        const float* xbb = x + (size_t)(b * C_IN + cc) * HWd;
#pragma unroll
        for (int i = 0; i < 4; ++i) {
            int j = i * 8 + wv;            // channel offset 0..31
            const float* xb = xbb + (size_t)j * HWd;
            float v = pq[0] * xb[po[0]] + pq[1] * xb[po[1]]
                    + pq[2] * xb[po[2]] + pq[3] * xb[po[3]];
            As[buf][mpix][j] = v;
        }
    };

    v8f acc[4];
    for (int n = 0; n < 4; ++n)
        for (int e = 0; e < 8; ++e) acc[n][e] = 0.0f;

    __syncthreads();           // params LDS ready
    loadParams(0);
    stageB(0, 0);
    buildA(0, 0);
    async_wait0();
    __syncthreads();

    for (int ch = 0; ch < NCHUNKS; ++ch) {
        const int buf = ch & 1;
        if (ch + 1 < NCHUNKS) {            // overlap next tile with WMMA
            if (((ch + 1) & 7) == 0) loadParams((ch + 1) >> 3);
            stageB(ch + 1, buf ^ 1);
            buildA(ch + 1, buf ^ 1);
        }
        // ---- 8 K-steps of v_wmma_f32_16x16x4_f32, 4 N-subtiles each
#pragma unroll
        for (int kq = 0; kq < KCHUNK; kq += 4) {
            const int kb = kq + khalf;     // even
            v2f a = *(const v2f*)&As[buf][Ms + lrow][kb];
#pragma unroll
            for (int ng = 0; ng < 4; ++ng) {
                v2f bf = *(const v2f*)&Bs[buf][kb >> 1][Nw + ng * 16 + lrow][0];
                acc[ng] = __builtin_amdgcn_wmma_f32_16x16x4_f32(
                    false, a, false, bf, (short)0, acc[ng], false, false);
            }
        }
        async_wait0();
        __syncthreads();
    }

    // ---- epilogue: transpose tile through LDS (reuse Bs), coalesced b128 stores
    float* Cs = &Bs[0][0][0][0];                 // 256 rows x CPAD floats = 36 KB
    const int Mbase = Ms + ((lane >> 4) << 3);   // +8 for lanes 16..31
#pragma unroll
    for (int ng = 0; ng < 4; ++ng) {
        int o = Nw + ng * 16 + lrow;
        float* row = Cs + o * CPAD + Mbase;
        *(float4*)(row)     = make_float4(acc[ng][0], acc[ng][1], acc[ng][2], acc[ng][3]);
        *(float4*)(row + 4) = make_float4(acc[ng][4], acc[ng][5], acc[ng][6], acc[ng][7]);
    }
    __syncthreads();
    {
        int o = tid;                             // one output channel per thread
        const float4* row = (const float4*)(Cs + o * CPAD);
        float4* op = (float4*)(out + (size_t)(b * O_OUT + o) * HWd + hw0);
#pragma unroll
        for (int r = 0; r < 8; ++r)
            op[r] = row[r];
    }
}

// ---------------- kernel 2: GroupNorm stats ----------------
__global__ __launch_bounds__(256) void gn_stats_k(const float* __restrict__ out,
                                                  float* __restrict__ stats) {
    const int bg = blockIdx.x;                 // b*8 + g
    const float* base = out + (size_t)bg * (32 * HWd);   // 32 contiguous channels
    float s = 0.0f, ss = 0.0f;
    for (int i = threadIdx.x; i < 32 * HWd; i += 256) {
        float v = base[i];
        s += v;
        ss += v * v;
    }
    __shared__ float rs[256], rq[256];
    rs[threadIdx.x] = s;
    rq[threadIdx.x] = ss;
    __syncthreads();
    for (int off = 128; off > 0; off >>= 1) {
        if (threadIdx.x < off) {
            rs[threadIdx.x] += rs[threadIdx.x + off];
            rq[threadIdx.x] += rq[threadIdx.x + off];
        }
        __syncthreads();
    }
    if (threadIdx.x == 0) {
        const float n = (float)(32 * HWd);
        float mu = rs[0] / n;
        float var = rq[0] / n - mu * mu;
        stats[bg * 2 + 0] = mu;
        stats[bg * 2 + 1] = rsqrtf(var + EPSV);
    }
}

// ---------------- kernel 3: normalize + affine + ReLU (in place) ----------------
__global__ __launch_bounds__(256) void gn_apply_k(float* __restrict__ out,
                                                  const float* __restrict__ stats,
                                                  const float* __restrict__ gamma,
                                                  const float* __restrict__ beta) {
    int idx = blockIdx.x * 256 + threadIdx.x;  // 0 .. B*O*HW-1
    float v = out[idx];
    int o  = (idx >> 12) & 255;
    int bg = idx >> 17;                        // idx / (32*4096) == b*8+g
    float mu = stats[bg * 2 + 0];
    float is = stats[bg * 2 + 1];
    float r = (v - mu) * is * gamma[o] + beta[o];
    out[idx] = fmaxf(r, 0.0f);
}

extern "C" void kernel_launch(void* const* d_in, const int* in_sizes, int n_in,
                              void* d_out, int out_size, void* d_ws, size_t ws_size,
                              hipStream_t stream) {
    const float* x      = (const float*)d_in[0];
    const float* angle  = (const float*)d_in[1];
    const float* mask   = (const float*)d_in[2];
    const float* weight = (const float*)d_in[3];
    const float* gamma  = (const float*)d_in[4];
    const float* beta   = (const float*)d_in[5];
    float* out = (float*)d_out;
    float* wTp   = (float*)d_ws;                          // 2304*256 fp32 = 2.36 MB
    float* stats = (float*)d_ws + (KK * C_IN * O_OUT);    // 64 fp32

    wt_transpose_k<<<KK * C_IN, 256, 0, stream>>>(weight, wTp);
    conv_wmma_k<<<(4 * HWd) / MTILE, 256, 0, stream>>>(x, angle, mask, wTp, out);
    gn_stats_k<<<4 * GROUPS, 256, 0, stream>>>(out, stats);
    gn_apply_k<<<(4 * O_OUT * HWd) / 256, 256, 0, stream>>>(out, stats, gamma, beta);
}